// PlackettLuceLoss_33612414058498
// MI455X (gfx1250) — compile-verified
//
#include <hip/hip_runtime.h>
#include <stdint.h>

#define H      18
#define BLOCK  256
#define TILE_ELEMS (BLOCK * H)   // 4608 elements per tile

typedef unsigned int u32x4 __attribute__((ext_vector_type(4)));
typedef int          i32x8 __attribute__((ext_vector_type(8)));
typedef int          i32x4 __attribute__((ext_vector_type(4)));

#if __has_builtin(__builtin_amdgcn_tensor_load_to_lds)
#define HAVE_TDM 1
#else
#define HAVE_TDM 0
#endif

// -----------------------------------------------------------------------------
// Build a 1-D contiguous TDM descriptor (tile == tensor chunk) and issue the
// DMA: global -> LDS.  data_size code: 0=1B, 1=2B, 2=4B, 3=8B.
// D# layout per CDNA5 ISA §8 (group0: count/lds_addr/global_addr/type,
// group1: data_size, tensor_dim0, tile_dim0, dim0_stride).
// Toolchain here is the 6-arg builtin variant (clang-23 / therock-10.0):
//   (u32x4 g0, i32x8 g1, i32x4 g2, i32x4 g3, i32x8 extra, i32 cpol)
// -----------------------------------------------------------------------------
#if HAVE_TDM
__device__ __forceinline__ void tdm_load_1d(uint32_t lds_off, const void* gptr,
                                            uint32_t n_elems, uint32_t dsize_code) {
  uint64_t ga = (uint64_t)(uintptr_t)gptr;
  u32x4 g0;
  g0[0] = 1u;                                              // count=1, is_restore=0, gather=0
  g0[1] = lds_off;                                         // LDS byte address
  g0[2] = (uint32_t)ga;                                    // global_addr[31:0]
  g0[3] = ((uint32_t)(ga >> 32) & 0x01FFFFFFu) | (2u << 30); // global_addr[56:32] | type=2
  i32x8 g1;
  g1[0] = (int)(dsize_code << 16);                         // wg_mask=0, data_size
  g1[1] = (int)((n_elems & 0xFFFFu) << 16);                // tensor_dim0[15:0]
  g1[2] = (int)(((n_elems >> 16) & 0xFFFFu) | (1u << 16)); // tensor_dim0[31:16] | tensor_dim1=1
  g1[3] = (int)((n_elems & 0xFFFFu) << 16);                // tile_dim0 = n_elems (1-D tile)
  g1[4] = 0;                                               // tile_dim1=0 (unused), tile_dim2=0
  g1[5] = (int)n_elems;                                    // tensor_dim0_stride[31:0]
  g1[6] = 0;
  g1[7] = 0;
  i32x4 gz4 = {0, 0, 0, 0};                                // groups 2/3 unused (<=2-D tensor)
  i32x8 gz8 = {0, 0, 0, 0, 0, 0, 0, 0};                    // extra group unused
  __builtin_amdgcn_tensor_load_to_lds(g0, g1, gz4, gz4, gz8, 0);
}
#endif

// -----------------------------------------------------------------------------
// Main kernel: one row (race) per thread; 256 rows per block.
// -----------------------------------------------------------------------------
__global__ void __launch_bounds__(BLOCK)
pl_main(const float* __restrict__ scores,
        const int* __restrict__ rankings,
        const unsigned char* __restrict__ mask,
        double* __restrict__ bsum, int* __restrict__ bcnt, int B) {
  __shared__ __align__(16) float          sS[TILE_ELEMS];
  __shared__ __align__(16) int            sR[TILE_ELEMS];
  __shared__ __align__(16) unsigned char  sM[TILE_ELEMS];
  __shared__ float red_s[BLOCK];
  __shared__ int   red_c[BLOCK];

  const int t        = threadIdx.x;
  const int rowStart = blockIdx.x * BLOCK;
  int rows = B - rowStart; if (rows > BLOCK) rows = BLOCK;
  const uint32_t nel = (uint32_t)rows * H;

#if HAVE_TDM
  if (t < 32) {  // wave 0 issues the DMAs (wave-level ops, EXEC ignored)
    tdm_load_1d((uint32_t)(uintptr_t)(void*)sS, scores   + (size_t)rowStart * H, nel, 2u);
    tdm_load_1d((uint32_t)(uintptr_t)(void*)sR, rankings + (size_t)rowStart * H, nel, 2u);
    tdm_load_1d((uint32_t)(uintptr_t)(void*)sM, mask     + (size_t)rowStart * H, nel, 0u);
    __builtin_amdgcn_s_wait_tensorcnt(0);
  }
#else
  for (uint32_t i = t; i < nel; i += BLOCK) {
    sS[i] = scores  [(size_t)rowStart * H + i];
    sR[i] = rankings[(size_t)rowStart * H + i];
    sM[i] = mask    [(size_t)rowStart * H + i];
  }
#endif
  __syncthreads();

  float loss = 0.0f;
  int raceValid = 0;

  if (t < rows) {
    const int base = t * H;
    float s[H];
    int   k[H];
    int   n = 0;
#pragma unroll
    for (int j = 0; j < H; ++j) {
      float sc = sS[base + j];
      int   r  = sR[base + j];
      bool  v  = (sM[base + j] != 0) && (r >= 0);
      s[j] = fminf(fmaxf(sc, -10.0f), 10.0f);
      k[j] = v ? r : -1;                      // -1 marks invalid
      n += v ? 1 : 0;
    }

    if (n >= 2) {
      // Traverse valid horses in DESCENDING stable-sort order (max ranking
      // first, ties broken by larger original index == reversed stable sort).
      // Running L = logsumexp of the suffix seen so far.
      unsigned used = 0;
      float L = 0.0f;
      for (int i = 0; i < n; ++i) {
        int   bestKey = -1;
        int   bestIdx = 0;
        float bestS   = 0.0f;
#pragma unroll
        for (int j = 0; j < H; ++j) {
          bool cand = (k[j] >= 0) && (((used >> j) & 1u) == 0u);
          if (cand && (k[j] >= bestKey)) { bestKey = k[j]; bestIdx = j; bestS = s[j]; }
        }
        used |= (1u << bestIdx);
        if (i == 0) {
          L = bestS;                           // last pick: deterministic, no loss term
        } else {
          float m = fmaxf(bestS, L);
          float d = fabsf(bestS - L);
          L = m + log1pf(__expf(-d));          // logaddexp
          float p = __expf(bestS - L);         // pick probability at this position
          p = fminf(fmaxf(p, 1e-8f), 1.0f - 1e-8f);
          loss += -__logf(p);
        }
      }
      raceValid = 1;
    }
  }

  // Deterministic fixed-order block reduction.
  red_s[t] = loss;
  red_c[t] = raceValid;
  __syncthreads();
  for (int off = BLOCK / 2; off > 0; off >>= 1) {
    if (t < off) { red_s[t] += red_s[t + off]; red_c[t] += red_c[t + off]; }
    __syncthreads();
  }
  if (t == 0) {
    bsum[blockIdx.x] = (double)red_s[0];
    bcnt[blockIdx.x] = red_c[0];
  }
}

// -----------------------------------------------------------------------------
// Final deterministic reduction over per-block partials.
// -----------------------------------------------------------------------------
__global__ void __launch_bounds__(256)
pl_reduce(const double* __restrict__ bsum, const int* __restrict__ bcnt,
          int nb, float* __restrict__ out) {
  __shared__ double ssum[256];
  __shared__ double scnt[256];
  const int t = threadIdx.x;
  double s = 0.0, c = 0.0;
  for (int i = t; i < nb; i += 256) { s += bsum[i]; c += (double)bcnt[i]; }
  ssum[t] = s; scnt[t] = c;
  __syncthreads();
  for (int off = 128; off > 0; off >>= 1) {
    if (t < off) { ssum[t] += ssum[t + off]; scnt[t] += scnt[t + off]; }
    __syncthreads();
  }
  if (t == 0) out[0] = (scnt[0] > 0.0) ? (float)(ssum[0] / scnt[0]) : 0.0f;
}

// -----------------------------------------------------------------------------
// Launch wrapper.
// -----------------------------------------------------------------------------
extern "C" void kernel_launch(void* const* d_in, const int* in_sizes, int n_in,
                              void* d_out, int out_size, void* d_ws, size_t ws_size,
                              hipStream_t stream) {
  const float*         scores   = (const float*)d_in[0];
  const int*           rankings = (const int*)d_in[1];
  const unsigned char* msk      = (const unsigned char*)d_in[2];

  const int B  = in_sizes[0] / H;                 // 1,000,000
  const int nb = (B + BLOCK - 1) / BLOCK;         // 3907 blocks

  double* bsum = (double*)d_ws;
  int*    bcnt = (int*)((char*)d_ws + (size_t)nb * sizeof(double));

  pl_main<<<nb, BLOCK, 0, stream>>>(scores, rankings, msk, bsum, bcnt, B);
  pl_reduce<<<1, 256, 0, stream>>>(bsum, bcnt, nb, (float*)d_out);
}